// _TrainLevel_15891378995269
// MI455X (gfx1250) — compile-verified
//
#include <hip/hip_runtime.h>
#include <stdint.h>

typedef __attribute__((ext_vector_type(4))) float v4f;

#define HG_THREADS 256  // 8 wave32 waves per block

#if defined(__AMDGCN__) && defined(__gfx1250__) && \
    __has_builtin(__builtin_amdgcn_global_load_async_to_lds_b128)
#define HG_ASYNC 1
// Match the builtin's parameter types exactly (from hipcc's diagnostic):
//   param0: int __attribute__((vector_size(16))) addrspace(1)*
//   param1: int __attribute__((vector_size(16))) addrspace(3)*
typedef int hg_v4i __attribute__((vector_size(16)));
typedef __attribute__((address_space(1))) hg_v4i* hg_gptr;
typedef __attribute__((address_space(3))) hg_v4i* hg_lptr;
#else
#define HG_ASYNC 0
#endif

// Zero the access-flag region of d_out (harness poisons it with 0xAA).
__launch_bounds__(HG_THREADS)
__global__ void hg_zero_access(float* __restrict__ access, int n) {
  int i = (int)blockIdx.x * HG_THREADS + (int)threadIdx.x;
  if (i < n) access[i] = 0.0f;
}

__launch_bounds__(HG_THREADS)
__global__ void hg_interp(const float* __restrict__ pts,
                          const float* __restrict__ vf,     // [n,8] f32
                          const int* __restrict__ h2v,      // [2^21] voxel ids (-1 = empty)
                          float* __restrict__ out,          // [P,8] f32
                          float* __restrict__ access,       // [n] f32 flags
                          int P) {
  const int tid = (int)threadIdx.x;
  const int p = (int)blockIdx.x * HG_THREADS + tid;
#if HG_ASYNC
  // 8 rows x 2 v4f per thread = 256 B/thread -> 64 KB/block (320 KB LDS/WGP).
  __shared__ v4f smem[HG_THREADS * 16];
#endif
  if (p >= P) return;

  const float px = pts[3 * p + 0];
  const float py = pts[3 * p + 1];
  const float pz = pts[3 * p + 2];

  // q = (pt - smin) / RES, f32 exactly like the reference
  const float qx = (px - (-10.0f)) / 0.1f;
  const float qy = (py - (-10.0f)) / 0.1f;
  const float qz = (pz - (-2.0f)) / 0.1f;

  const float bx = floorf(qx), by = floorf(qy), bz = floorf(qz);
  const float fx = qx - bx, fy = qy - by, fz = qz - bz;

  // All cell indices are non-negative; BUCKETS = 2^21 is a power of two, so
  // the int64 (idx*primes).sum % BUCKETS equals wrapping-u32 math & mask.
  const uint32_t ix = (uint32_t)(int)bx;
  const uint32_t iy = (uint32_t)(int)by;
  const uint32_t iz = (uint32_t)(int)bz;

  const uint32_t kP1 = 2654435761u, kP2 = 805459861u;
  const uint32_t kMask = (1u << 21) - 1u;
  const uint32_t h000 = ix + iy * kP1 + iz * kP2;

  int vid[8];
#pragma unroll
  for (int c = 0; c < 8; ++c) {
    const uint32_t hc = (h000 + (uint32_t)(c & 1) +
                         (uint32_t)((c >> 1) & 1) * kP1 +
                         (uint32_t)(c >> 2) * kP2) & kMask;
    vid[c] = h2v[hc];  // 8 independent table gathers issued together
  }

  const float gx = 1.0f - fx, gy = 1.0f - fy, gz = 1.0f - fz;
  const float wxv[2] = {gx, fx};
  const float wyv[2] = {gy, fy};
  const float wzv[2] = {gz, fz};

  float w8[8];
#pragma unroll
  for (int c = 0; c < 8; ++c) {
    const int v = vid[c];
    float w = wxv[c & 1] * wyv[(c >> 1) & 1] * wzv[c >> 2];
    if (v >= 0) {
      access[v] = 1.0f;  // idempotent scatter; only True is ever written
    } else {
      w = 0.0f;          // masked corner: weight 0 x finite feat == 0 (== ref)
      vid[c] = 0;        // safe gather address
    }
    w8[c] = w;
  }

  v4f acc0 = {0.0f, 0.0f, 0.0f, 0.0f};
  v4f acc1 = {0.0f, 0.0f, 0.0f, 0.0f};

#if HG_ASYNC
  // CDNA5 async gather: per-lane global addresses -> per-lane LDS slots,
  // tracked on ASYNCcnt. 16 gathers in flight per lane, single wait.
  {
    const uint32_t dbase = (uint32_t)(uintptr_t)&smem[tid * 16];
#pragma unroll
    for (int c = 0; c < 8; ++c) {
      const float* src = vf + (size_t)vid[c] * 8;  // 32B-aligned row
      const uint32_t dlo = dbase + (uint32_t)(c * 32);
      __builtin_amdgcn_global_load_async_to_lds_b128(
          (hg_gptr)(uintptr_t)src, (hg_lptr)dlo, 0, 0);
      __builtin_amdgcn_global_load_async_to_lds_b128(
          (hg_gptr)(uintptr_t)(src + 4), (hg_lptr)(dlo + 16u), 0, 0);
    }
#if __has_builtin(__builtin_amdgcn_s_wait_asynccnt)
    __builtin_amdgcn_s_wait_asynccnt(0);
#else
    asm volatile("s_wait_asynccnt 0" ::: "memory");
#endif
#pragma unroll
    for (int c = 0; c < 8; ++c) {
      const v4f lo = smem[tid * 16 + c * 2 + 0];
      const v4f hi = smem[tid * 16 + c * 2 + 1];
      acc0 += w8[c] * lo;
      acc1 += w8[c] * hi;
    }
  }
#else
  // Fallback: direct b128 gathers into VGPRs, all 16 issued before use.
#pragma unroll
  for (int c = 0; c < 8; ++c) {
    const v4f* fp = (const v4f*)(vf + (size_t)vid[c] * 8);  // 32B-aligned rows
    const v4f lo = fp[0];
    const v4f hi = fp[1];
    acc0 += w8[c] * lo;
    acc1 += w8[c] * hi;
  }
#endif

  // Streamed output: nontemporal so the 32 MB result doesn't evict the
  // L2-resident hash table + feature table.
  v4f* op = (v4f*)(out + (size_t)p * 8);
  __builtin_nontemporal_store(acc0, op);
  __builtin_nontemporal_store(acc1, op + 1);
}

extern "C" void kernel_launch(void* const* d_in, const int* in_sizes, int n_in,
                              void* d_out, int out_size, void* d_ws, size_t ws_size,
                              hipStream_t stream) {
  (void)n_in; (void)out_size; (void)d_ws; (void)ws_size;
  const float* pts = (const float*)d_in[0];
  const float* vf  = (const float*)d_in[1];
  const int*   h2v = (const int*)d_in[2];
  float* out = (float*)d_out;

  const int P = in_sizes[0] / 3;   // 1,000,000 points
  const int n = in_sizes[1] / 8;   // 1,600,000 voxels (D = 8)
  float* access = out + (size_t)P * 8;

  hg_zero_access<<<(n + HG_THREADS - 1) / HG_THREADS, HG_THREADS, 0, stream>>>(access, n);
  hg_interp<<<(P + HG_THREADS - 1) / HG_THREADS, HG_THREADS, 0, stream>>>(
      pts, vf, h2v, out, access, P);
}